// SEBlock_14482629722931
// MI455X (gfx1250) — compile-verified
//
#include <hip/hip_runtime.h>

typedef __attribute__((ext_vector_type(2))) float v2f;
typedef __attribute__((ext_vector_type(8))) float v8f;

#define B_DIM   32
#define C_DIM   512
#define CR_DIM  32
#define HW      3136   // 56*56
#define HW4     784    // HW/4 float4s per (b,c) plane

// ---------------------------------------------------------------------------
// Phase 1: squeeze — per-(b,c) spatial mean. One 256-thread block per (b,c)
// pair; fully coalesced float4 loads, LDS tree reduction.
// ---------------------------------------------------------------------------
__global__ __launch_bounds__(256) void se_squeeze_kernel(
    const float4* __restrict__ x, float* __restrict__ sq) {
    const int p = blockIdx.x;               // (b*C + c) in [0, 16384)
    const float4* base = x + (size_t)p * HW4;
    float sum = 0.0f;
    for (int t = threadIdx.x; t < HW4; t += 256) {
        float4 v = base[t];
        sum += v.x + v.y + v.z + v.w;
    }
    __shared__ float red[256];
    red[threadIdx.x] = sum;
    __syncthreads();
    for (int s = 128; s > 0; s >>= 1) {
        if (threadIdx.x < s) red[threadIdx.x] += red[threadIdx.x + s];
        __syncthreads();
    }
    if (threadIdx.x == 0) sq[p] = red[0] * (1.0f / (float)HW);
}

// ---------------------------------------------------------------------------
// Phase 2: excitation — sigmoid(relu(S*W1^T + b1)*W2^T + b2), all in one
// 128-thread workgroup using V_WMMA_F32_16X16X4_F32 (exact f32 math).
//
// Fragment layouts per CDNA5 ISA 7.12.2 (wave32):
//   A 16x4 f32 : lane l<16 -> row M=l, holds K=0,1 ; lane l>=16 -> row M=l-16,
//                holds K=2,3  => per-lane float2 at [m][k0 + 2*(lane>>4)]
//   B 4x16 f32 : mirrored, lane gives column N, float2 over K
//   D 16x16    : N = lane&15, VGPR r holds M = r + 8*(lane>>4)
// ---------------------------------------------------------------------------
__global__ __launch_bounds__(128) void se_excite_kernel(
    const float* __restrict__ sq,     // [32][512]
    const float* __restrict__ w1,     // [32][512]
    const float* __restrict__ bias1,  // [32]
    const float* __restrict__ w2,     // [512][32]
    const float* __restrict__ bias2,  // [512]
    float* __restrict__ e_out) {      // [32][512]
    __shared__ float e1[B_DIM * CR_DIM];   // relu(GEMM1) result, 32x32

    const int tid  = threadIdx.x;
    const int wave = tid >> 5;   // 0..3
    const int lane = tid & 31;
    const int half = lane >> 4;  // 0: K={0,1}, 1: K={2,3}
    const int l    = lane & 15;

    // ---- GEMM1: e1[32,32] = S[32,512] x w1^T[512,32] ; one tile per wave ----
    {
        const int mt = wave >> 1, nt = wave & 1;
        v8f acc = {};
        const float* arow = sq + (size_t)(mt * 16 + l) * C_DIM + 2 * half;
        const float* brow = w1 + (size_t)(nt * 16 + l) * C_DIM + 2 * half;
        for (int k0 = 0; k0 < C_DIM; k0 += 4) {
            v2f a = *(const v2f*)(arow + k0);
            v2f b = *(const v2f*)(brow + k0);
            acc = __builtin_amdgcn_wmma_f32_16x16x4_f32(
                false, a, false, b, (short)0, acc, false, false);
        }
        const int n = nt * 16 + l;
        const float bb = bias1[n];
#pragma unroll
        for (int r = 0; r < 8; ++r) {
            const int m = mt * 16 + r + 8 * half;
            e1[m * CR_DIM + n] = fmaxf(acc[r] + bb, 0.0f);
        }
    }
    __syncthreads();

    // ---- GEMM2: e[32,512] = e1[32,32] x w2^T[32,512] ; 16 tiles per wave ----
    for (int nt2 = wave; nt2 < 32; nt2 += 4) {       // uniform across waves
#pragma unroll
        for (int mt2 = 0; mt2 < 2; ++mt2) {
            v8f acc = {};
            const float* arow = &e1[(mt2 * 16 + l) * CR_DIM + 2 * half];
            const float* brow = w2 + (size_t)(nt2 * 16 + l) * CR_DIM + 2 * half;
#pragma unroll
            for (int k0 = 0; k0 < CR_DIM; k0 += 4) {
                v2f a = *(const v2f*)(arow + k0);
                v2f b = *(const v2f*)(brow + k0);
                acc = __builtin_amdgcn_wmma_f32_16x16x4_f32(
                    false, a, false, b, (short)0, acc, false, false);
            }
            const int n = nt2 * 16 + l;
            const float bb = bias2[n];
#pragma unroll
            for (int r = 0; r < 8; ++r) {
                const int m = mt2 * 16 + r + 8 * half;
                const float v = acc[r] + bb;
                e_out[m * C_DIM + n] = 1.0f / (1.0f + __expf(-v));
            }
        }
    }
}

// ---------------------------------------------------------------------------
// Phase 3: scale — out = x * e[b][c], float4 coalesced (bandwidth-bound).
// ---------------------------------------------------------------------------
__global__ __launch_bounds__(256) void se_scale_kernel(
    const float4* __restrict__ x, const float* __restrict__ e,
    float4* __restrict__ out, int total4) {
    const int i = blockIdx.x * 256 + threadIdx.x;
    if (i >= total4) return;
    const float s = e[i / HW4];     // flat (b*C + c) index
    float4 v = x[i];
    v.x *= s; v.y *= s; v.z *= s; v.w *= s;
    out[i] = v;
}

// ---------------------------------------------------------------------------
extern "C" void kernel_launch(void* const* d_in, const int* in_sizes, int n_in,
                              void* d_out, int out_size, void* d_ws, size_t ws_size,
                              hipStream_t stream) {
    const float* x     = (const float*)d_in[0];   // [32,512,56,56]
    const float* w1    = (const float*)d_in[1];   // [32,512]
    const float* bias1 = (const float*)d_in[2];   // [32]
    const float* w2    = (const float*)d_in[3];   // [512,32]
    const float* bias2 = (const float*)d_in[4];   // [512]
    float* out = (float*)d_out;

    float* sq = (float*)d_ws;                 // [32*512] squeeze means
    float* e  = sq + B_DIM * C_DIM;           // [32*512] sigmoid scales

    se_squeeze_kernel<<<B_DIM * C_DIM, 256, 0, stream>>>((const float4*)x, sq);
    se_excite_kernel<<<1, 128, 0, stream>>>(sq, w1, bias1, w2, bias2, e);

    const int total4 = B_DIM * C_DIM * HW4;
    se_scale_kernel<<<(total4 + 255) / 256, 256, 0, stream>>>(
        (const float4*)x, e, (float4*)out, total4);
}